// DCGRUCell_60533269069993
// MI455X (gfx1250) — compile-verified
//
#include <hip/hip_runtime.h>
#include <hip/hip_bf16.h>

typedef __attribute__((ext_vector_type(2))) float v2f;
typedef __attribute__((ext_vector_type(8))) float v8f;

#define NN   50000
#define NE   640000
#define H    64
#define H2   128

// Non-returning hardware float atomic add (global_atomic_add_f32, STOREcnt).
__device__ __forceinline__ void atomic_fadd(float* p, float v) {
    asm volatile("global_atomic_add_f32 %0, %1, off"
                 :: "v"(p), "v"(v)
                 : "memory");
}

// Async copy one dword from global memory directly into LDS (ASYNCcnt).
__device__ __forceinline__ void async_g2lds_b32(const float* g, const float* lds_generic) {
    unsigned lds_off = (unsigned)(uintptr_t)lds_generic;  // low 32 bits = LDS byte offset
    asm volatile("global_load_async_to_lds_b32 %0, %1, off"
                 :: "v"(lds_off), "v"(g)
                 : "memory");
}

__device__ __forceinline__ void wait_async_all() {
    asm volatile("s_wait_asynccnt 0" ::: "memory");
}

// ---------------------------------------------------------------------------
// zero-fill
// ---------------------------------------------------------------------------
__global__ void fill_zero(float* __restrict__ p, long n) {
    long stride = (long)gridDim.x * blockDim.x;
    for (long t = (long)blockIdx.x * blockDim.x + threadIdx.x; t < n; t += stride)
        p[t] = 0.0f;
}

// ---------------------------------------------------------------------------
// edge scatter #1: aggF[dst] += w*feat[src], aggS[dst] += w*state[src]
// one thread handles a float2 slice of one edge  (E * 32 work items)
// ---------------------------------------------------------------------------
__global__ void scatter_fs(const float* __restrict__ feat,
                           const float* __restrict__ state,
                           const int*   __restrict__ src,
                           const int*   __restrict__ dst,
                           const float* __restrict__ ew,
                           float* __restrict__ aggF,
                           float* __restrict__ aggS) {
    const long total  = (long)NE * 32;
    const long stride = (long)gridDim.x * blockDim.x;
    for (long t = (long)blockIdx.x * blockDim.x + threadIdx.x; t < total; t += stride) {
        int e = (int)(t >> 5);
        int p = ((int)t & 31) * 2;
        int u = src[e], v = dst[e];
        float w = ew[e];
        v2f f = *(const v2f*)(feat  + (size_t)u * H + p);
        v2f s = *(const v2f*)(state + (size_t)u * H + p);
        float* dF = aggF + (size_t)v * H + p;
        float* dS = aggS + (size_t)v * H + p;
        atomic_fadd(dF + 0, w * f.x);
        atomic_fadd(dF + 1, w * f.y);
        atomic_fadd(dS + 0, w * s.x);
        atomic_fadd(dS + 1, w * s.y);
    }
}

// ---------------------------------------------------------------------------
// edge scatter #2: aggRS[dst] += w * rs[src]
// ---------------------------------------------------------------------------
__global__ void scatter_rs(const float* __restrict__ rs,
                           const int*   __restrict__ src,
                           const int*   __restrict__ dst,
                           const float* __restrict__ ew,
                           float* __restrict__ aggRS) {
    const long total  = (long)NE * 32;
    const long stride = (long)gridDim.x * blockDim.x;
    for (long t = (long)blockIdx.x * blockDim.x + threadIdx.x; t < total; t += stride) {
        int e = (int)(t >> 5);
        int p = ((int)t & 31) * 2;
        int u = src[e], v = dst[e];
        float w = ew[e];
        v2f r = *(const v2f*)(rs + (size_t)u * H + p);
        float* d = aggRS + (size_t)v * H + p;
        atomic_fadd(d + 0, w * r.x);
        atomic_fadd(d + 1, w * r.y);
    }
}

// ---------------------------------------------------------------------------
// GEMM1: zr = sigmoid([aggF|aggS] @ Wzr + bzr)   (Wzr: [128,128] row-major)
// fused: zbuf = zr[:, :64];  rsbuf = zr[:, 64:] * state
// 8 waves / block, each wave: 16 rows x 128 cols via V_WMMA_F32_16X16X4_F32
// ---------------------------------------------------------------------------
__global__ __launch_bounds__(256) void gemm_zr(const float* __restrict__ aggF,
                                               const float* __restrict__ aggS,
                                               const float* __restrict__ Wzr,
                                               const float* __restrict__ bzr,
                                               const float* __restrict__ state,
                                               float* __restrict__ zbuf,
                                               float* __restrict__ rsbuf) {
    __shared__ float lW[H2 * H2];   // 64 KB
    __shared__ float lB[H2];
    // async global -> LDS staging of Wzr (no VGPR bounce)
    for (int i = threadIdx.x; i < H2 * H2; i += 256)
        async_g2lds_b32(Wzr + i, &lW[i]);
    if (threadIdx.x < H2) lB[threadIdx.x] = bzr[threadIdx.x];
    wait_async_all();
    __syncthreads();

    const int wave = threadIdx.x >> 5;
    const int lane = threadIdx.x & 31;
    const int half = lane >> 4;       // 0: lanes 0-15, 1: lanes 16-31
    const int ln   = lane & 15;
    const int rowBase = (blockIdx.x * 8 + wave) * 16;
    if (rowBase >= NN) return;        // wave-uniform exit (after barrier)

    int arow = rowBase + ln;
    if (arow >= NN) arow = NN - 1;    // clamp for tail; stores masked later
    const float* A0 = aggF + (size_t)arow * H;
    const float* A1 = aggS + (size_t)arow * H;

    v8f acc[8];
#pragma unroll
    for (int j = 0; j < 8; ++j) acc[j] = (v8f){0,0,0,0,0,0,0,0};

    // K = 0..63 from aggF
#pragma unroll 4
    for (int kk = 0; kk < 16; ++kk) {
        const int kc = kk * 4 + 2 * half;              // lane-half K offset
        v2f a = *(const v2f*)(A0 + kc);
#pragma unroll
        for (int j = 0; j < 8; ++j) {
            v2f b;
            b.x = lW[(kc    ) * H2 + j * 16 + ln];
            b.y = lW[(kc + 1) * H2 + j * 16 + ln];
            acc[j] = __builtin_amdgcn_wmma_f32_16x16x4_f32(
                false, a, false, b, (short)0, acc[j], false, false);
        }
    }
    // K = 64..127 from aggS
#pragma unroll 4
    for (int kk = 0; kk < 16; ++kk) {
        const int kc = kk * 4 + 2 * half;
        const int kW = H + kc;
        v2f a = *(const v2f*)(A1 + kc);
#pragma unroll
        for (int j = 0; j < 8; ++j) {
            v2f b;
            b.x = lW[(kW    ) * H2 + j * 16 + ln];
            b.y = lW[(kW + 1) * H2 + j * 16 + ln];
            acc[j] = __builtin_amdgcn_wmma_f32_16x16x4_f32(
                false, a, false, b, (short)0, acc[j], false, false);
        }
    }

    // epilogue: bias + sigmoid, split z / r, store z and r*state
#pragma unroll
    for (int j = 0; j < 8; ++j) {
        const int col  = j * 16 + ln;
        const float bias = lB[col];
#pragma unroll
        for (int i = 0; i < 8; ++i) {
            const int m = rowBase + i + 8 * half;      // C layout: M=i / M=i+8
            if (m < NN) {
                float v = acc[j][i] + bias;
                float s = 1.0f / (1.0f + __expf(-v));
                if (col < H) {
                    zbuf[(size_t)m * H + col] = s;                     // z
                } else {
                    float st = state[(size_t)m * H + (col - H)];
                    rsbuf[(size_t)m * H + (col - H)] = s * st;         // r*state
                }
            }
        }
    }
}

// ---------------------------------------------------------------------------
// GEMM2: c = tanh([aggF|aggRS] @ Wc + bc)   (Wc: [128,64] row-major)
// fused: out = z*state + (1-z)*c
// ---------------------------------------------------------------------------
__global__ __launch_bounds__(256) void gemm_c(const float* __restrict__ aggF,
                                              const float* __restrict__ aggRS,
                                              const float* __restrict__ Wc,
                                              const float* __restrict__ bc,
                                              const float* __restrict__ state,
                                              const float* __restrict__ zbuf,
                                              float* __restrict__ out) {
    __shared__ float lW[H2 * H];    // 32 KB
    __shared__ float lB[H];
    for (int i = threadIdx.x; i < H2 * H; i += 256)
        async_g2lds_b32(Wc + i, &lW[i]);
    if (threadIdx.x < H) lB[threadIdx.x] = bc[threadIdx.x];
    wait_async_all();
    __syncthreads();

    const int wave = threadIdx.x >> 5;
    const int lane = threadIdx.x & 31;
    const int half = lane >> 4;
    const int ln   = lane & 15;
    const int rowBase = (blockIdx.x * 8 + wave) * 16;
    if (rowBase >= NN) return;

    int arow = rowBase + ln;
    if (arow >= NN) arow = NN - 1;
    const float* A0 = aggF  + (size_t)arow * H;
    const float* A1 = aggRS + (size_t)arow * H;

    v8f acc[4];
#pragma unroll
    for (int j = 0; j < 4; ++j) acc[j] = (v8f){0,0,0,0,0,0,0,0};

#pragma unroll 4
    for (int kk = 0; kk < 16; ++kk) {
        const int kc = kk * 4 + 2 * half;
        v2f a = *(const v2f*)(A0 + kc);
#pragma unroll
        for (int j = 0; j < 4; ++j) {
            v2f b;
            b.x = lW[(kc    ) * H + j * 16 + ln];
            b.y = lW[(kc + 1) * H + j * 16 + ln];
            acc[j] = __builtin_amdgcn_wmma_f32_16x16x4_f32(
                false, a, false, b, (short)0, acc[j], false, false);
        }
    }
#pragma unroll 4
    for (int kk = 0; kk < 16; ++kk) {
        const int kc = kk * 4 + 2 * half;
        const int kW = H + kc;
        v2f a = *(const v2f*)(A1 + kc);
#pragma unroll
        for (int j = 0; j < 4; ++j) {
            v2f b;
            b.x = lW[(kW    ) * H + j * 16 + ln];
            b.y = lW[(kW + 1) * H + j * 16 + ln];
            acc[j] = __builtin_amdgcn_wmma_f32_16x16x4_f32(
                false, a, false, b, (short)0, acc[j], false, false);
        }
    }

    // epilogue: tanh + GRU gate combine
#pragma unroll
    for (int j = 0; j < 4; ++j) {
        const int col  = j * 16 + ln;
        const float bias = lB[col];
#pragma unroll
        for (int i = 0; i < 8; ++i) {
            const int m = rowBase + i + 8 * half;
            if (m < NN) {
                float c  = tanhf(acc[j][i] + bias);
                float z  = zbuf [(size_t)m * H + col];
                float st = state[(size_t)m * H + col];
                out[(size_t)m * H + col] = z * st + (1.0f - z) * c;
            }
        }
    }
}

// ---------------------------------------------------------------------------
// host-side orchestration
// ---------------------------------------------------------------------------
extern "C" void kernel_launch(void* const* d_in, const int* in_sizes, int n_in,
                              void* d_out, int out_size, void* d_ws, size_t ws_size,
                              hipStream_t stream) {
    const float* feat  = (const float*)d_in[0];   // [N,64]
    const float* state = (const float*)d_in[1];   // [N,64]
    const int*   src   = (const int*)  d_in[2];   // [E]
    const int*   dst   = (const int*)  d_in[3];   // [E]
    const float* ew    = (const float*)d_in[4];   // [E]
    const float* Wzr   = (const float*)d_in[5];   // [128,128]
    const float* bzr   = (const float*)d_in[6];   // [128]
    const float* Wc    = (const float*)d_in[7];   // [128,64]
    const float* bc    = (const float*)d_in[8];   // [64]
    float*       out   = (float*)d_out;           // [N,64]

    float* ws    = (float*)d_ws;
    float* aggF  = ws;                       // [N,64]
    float* aggS  = ws + (size_t)NN * H;      // [N,64]  (reused as aggRS)
    float* zbuf  = ws + (size_t)2 * NN * H;  // [N,64]
    float* rsbuf = ws + (size_t)3 * NN * H;  // [N,64]

    const int TB = 256;
    const int scatterBlocks = 4096;
    const int gemmBlocks    = (NN + 127) / 128;   // 8 waves x 16 rows per block

    // 1) zero aggF|aggS (contiguous)
    fill_zero<<<2048, TB, 0, stream>>>(aggF, (long)2 * NN * H);
    // 2) weighted scatter of feat & state
    scatter_fs<<<scatterBlocks, TB, 0, stream>>>(feat, state, src, dst, ew, aggF, aggS);
    // 3) GEMM1 + sigmoid -> z, r*state
    gemm_zr<<<gemmBlocks, TB, 0, stream>>>(aggF, aggS, Wzr, bzr, state, zbuf, rsbuf);
    // 4) zero aggRS (reuse aggS slab)
    fill_zero<<<2048, TB, 0, stream>>>(aggS, (long)NN * H);
    // 5) weighted scatter of r*state
    scatter_rs<<<scatterBlocks, TB, 0, stream>>>(rsbuf, src, dst, ew, aggS);
    // 6) GEMM2 + tanh + gate combine -> out
    gemm_c<<<gemmBlocks, TB, 0, stream>>>(aggF, aggS, Wc, bc, state, zbuf, out);
}